// IGSC_3D_59700045415095
// MI455X (gfx1250) — compile-verified
//
#include <hip/hip_runtime.h>
#include <hip/hip_bf16.h>

typedef float v2f __attribute__((ext_vector_type(2)));
typedef float v8f __attribute__((ext_vector_type(8)));

#define F_IN   128
#define CCH    128
#define NPTS   40000
#define NBATCH 4
#define VOXELS 4096            // 16*16*16
#define VOL_B_STRIDE (CCH * VOXELS)  // 524288 floats per batch
#define OUT_ROW 259            // 128 + 128 + 3

// One-time volume transpose: [B,C,4096] -> [B,4096,C] (coalesced-gather layout)
__global__ void IGSC_vol_transpose(const float* __restrict__ vol,
                                   float* __restrict__ volT, int total) {
    int t = blockIdx.x * blockDim.x + threadIdx.x;
    if (t >= total) return;
    int c = t & (CCH - 1);
    int v = (t >> 7) & (VOXELS - 1);
    int b = t >> 19;
    volT[t] = vol[(size_t)(b * CCH + c) * VOXELS + v];
}

// Fused: positions (WMMA f32 16x16x4) + trilinear sample + concat writes.
// vol strides parameterized: transposed (ch=1, vox=128) or original (ch=4096, vox=1).
__global__ void __launch_bounds__(256)
IGSC_fused(const float* __restrict__ x, const float* __restrict__ W,
           const float* __restrict__ vol, float* __restrict__ out,
           float* __restrict__ pos_out, int ch_stride, int vox_stride) {
    __shared__ float Wlds[16 * F_IN];        // [n][f], cols >=3 zero-padded
    __shared__ float pos_lds[8][16][4];      // [wave][point-in-tile][xyz]

    const int tid  = threadIdx.x;
    const int wave = tid >> 5;
    const int lane = tid & 31;
    const int m    = lane & 15;
    const int half = lane >> 4;

    // Stage W transposed+padded into LDS
    for (int e = tid; e < 16 * F_IN; e += 256) {
        int n = e >> 7, f = e & (F_IN - 1);
        Wlds[e] = (n < 3) ? W[f * 3 + n] : 0.0f;
    }
    __syncthreads();

    const int tile  = blockIdx.x * 8 + wave;   // 1250*8 = 10000 tiles exactly
    const int pbase = tile * 16;               // 16 points per tile
    const int b     = pbase / NPTS;            // tile never crosses batch (40000%16==0)

    // ---- positions = x @ W via V_WMMA_F32_16X16X4_F32, K=128 in 32 steps ----
    const float* xrow = x + (size_t)(pbase + m) * F_IN + (half << 1);
    const float* wrow = &Wlds[m * F_IN + (half << 1)];
    v8f acc = {0.f, 0.f, 0.f, 0.f, 0.f, 0.f, 0.f, 0.f};
#pragma unroll
    for (int kb = 0; kb < F_IN; kb += 4) {
        v2f a  = *(const v2f*)(xrow + kb);   // A[m][kb + half*2 + v]
        v2f bw = *(const v2f*)(wrow + kb);   // B[kb + half*2 + v][n=m]
        acc = __builtin_amdgcn_wmma_f32_16x16x4_f32(
            false, a, false, bw, (short)0, acc, false, false);
    }
    // C layout: acc[r] = D[r + 8*half][lane&15]; columns 0..2 are positions
    if (m < 3) {
#pragma unroll
        for (int r = 0; r < 8; ++r)
            pos_lds[wave][r + (half << 3)][m] = acc[r];
    }
    __syncthreads();

    // ---- per-point: copy x, write positions, trilinear-sample 128 channels ----
    const size_t vbase = (size_t)b * VOL_B_STRIDE;
    for (int r = 0; r < 16; ++r) {
        const int row = pbase + r;
        float* orow = out + (size_t)row * OUT_ROW;
        const float* xr = x + (size_t)row * F_IN;

        // x passthrough (coalesced b32) — tile is L0-hot from fragment loads
        for (int c = lane; c < F_IN; c += 32) orow[c] = xr[c];
        if (lane < 3) {
            float p = pos_lds[wave][r][lane];
            orow[256 + lane] = p;
            pos_out[(size_t)row * 3 + lane] = p;
        }

        const float px = pos_lds[wave][r][0];
        const float py = pos_lds[wave][r][1];
        const float pz = pos_lds[wave][r][2];
        // mirror reference math: grid = 2p-1; i = (g+1)*0.5*(dim-1)
        const float ix = ((2.0f * px - 1.0f) + 1.0f) * 0.5f * 15.0f;
        const float iy = ((2.0f * py - 1.0f) + 1.0f) * 0.5f * 15.0f;
        const float iz = ((2.0f * pz - 1.0f) + 1.0f) * 0.5f * 15.0f;
        const float x0f = floorf(ix), y0f = floorf(iy), z0f = floorf(iz);
        const float fx = ix - x0f, fy = iy - y0f, fz = iz - z0f;
        const int x0 = (int)x0f, y0 = (int)y0f, z0 = (int)z0f;
        const int xs[2] = {x0, x0 + 1}, ys[2] = {y0, y0 + 1}, zs[2] = {z0, z0 + 1};
        const float wxs[2] = {1.0f - fx, fx};
        const float wys[2] = {1.0f - fy, fy};
        const float wzs[2] = {1.0f - fz, fz};

        float wgt[8]; int off[8];
#pragma unroll
        for (int zi = 0; zi < 2; ++zi)
#pragma unroll
            for (int yi = 0; yi < 2; ++yi)
#pragma unroll
                for (int xi = 0; xi < 2; ++xi) {
                    const int k = (zi * 2 + yi) * 2 + xi;
                    const int xc = xs[xi], yc = ys[yi], zc = zs[zi];
                    const bool valid = (xc >= 0) & (xc < 16) & (yc >= 0) & (yc < 16) &
                                       (zc >= 0) & (zc < 16);
                    const int cx = min(max(xc, 0), 15);
                    const int cy = min(max(yc, 0), 15);
                    const int cz = min(max(zc, 0), 15);
                    off[k] = (cz * 16 + cy) * 16 + cx;
                    wgt[k] = wxs[xi] * wys[yi] * wzs[zi] * (valid ? 1.0f : 0.0f);
                }

        for (int c = lane; c < CCH; c += 32) {
            float s = 0.0f;
#pragma unroll
            for (int k = 0; k < 8; ++k)
                s += wgt[k] * vol[vbase + (size_t)off[k] * vox_stride +
                                  (size_t)c * ch_stride];
            orow[128 + c] = s;
        }
    }
}

extern "C" void kernel_launch(void* const* d_in, const int* in_sizes, int n_in,
                              void* d_out, int out_size, void* d_ws, size_t ws_size,
                              hipStream_t stream) {
    const float* x   = (const float*)d_in[0];
    // d_in[1] = adj (int64) — unused by the reference math
    const float* vol = (const float*)d_in[2];
    const float* W   = (const float*)d_in[3];

    float* out     = (float*)d_out;
    float* pos_out = out + (size_t)NBATCH * NPTS * OUT_ROW;

    const size_t volElems = (size_t)NBATCH * CCH * VOXELS;  // 2,097,152 floats
    const float* volp;
    int ch_stride, vox_stride;
    if (ws_size >= volElems * sizeof(float)) {
        IGSC_vol_transpose<<<(int)(volElems / 256), 256, 0, stream>>>(
            vol, (float*)d_ws, (int)volElems);
        volp = (const float*)d_ws;
        ch_stride = 1;
        vox_stride = CCH;       // coalesced channel reads
    } else {
        volp = vol;
        ch_stride = VOXELS;     // original strided layout fallback
        vox_stride = 1;
    }

    // 160000 points / 16 per wave = 10000 waves = 1250 blocks x 8 waves
    IGSC_fused<<<1250, 256, 0, stream>>>(x, W, volp, out, pos_out,
                                         ch_stride, vox_stride);
}